// IELin_31791347925862
// MI455X (gfx1250) — compile-verified
//
#include <hip/hip_runtime.h>
#include <hip/hip_bf16.h>

typedef __attribute__((ext_vector_type(2))) float v2f;
typedef __attribute__((ext_vector_type(8))) float v8f;

#define IN_DIM 1184
#define OUT_DIM 1184
#define ROWS 32
#define WAVES 8

// Interleaved LDS layout: logical row r lives at (r&15)*LDS_PAIR + (r>>4)*LDS_HALF.
// - sub1 sits only +1184 floats (+4736B) from sub0 -> all A loads reachable via
//   16-bit DS immediates from ONE per-wave base register (no address VALU).
// - lane stride across the 16 A-rows = 2372 words; 2372 mod 64 = 4 and 4*r mod 64
//   is distinct for r=0..15 -> conflict-free banks.
// - PAIR,HALF both multiples of 4 -> b128 staging stores and b64 A loads aligned.
#define LDS_HALF 1184
#define LDS_PAIR 2372
#define LDS_FLOATS (16 * LDS_PAIR)          // 37,952 floats
#define LDS_BYTES  (LDS_FLOATS * 4)         // 151,808 B -> 2 WGs per 320KB WGP

// One (32-row x 16-col) tile of one diagonal block: two 16-row accumulators
// sharing each B fragment (B is hoisted to VGPRs and reused by both).
// xsrow = &xs[arow*LDS_PAIR + koff]; wbase = W + bcol;
// obase = out + (row0 + half*8)*OUT_DIM + bcol
template <int K, bool TAIL>
__device__ __forceinline__ void do_tile(const float* __restrict__ xsrow,
                                        const float* __restrict__ wbase,
                                        float* __restrict__ obase,
                                        int in_off, int out_off, int woff,
                                        int koff, int rows_from_half) {
    const float* __restrict__ a0 = xsrow + in_off;
    const float* __restrict__ a1 = a0 + LDS_HALF;
    const float* __restrict__ wb = wbase + woff + koff * 128;

    v8f acc0 = {};
    v8f acc1 = {};
#pragma unroll
    for (int kk = 0; kk < K; kk += 4) {
        v2f b;
        b.x = wb[kk * 128];                      // global_load_b32 (L2-resident W)
        b.y = wb[kk * 128 + 128];
        v2f x0 = *(const v2f*)(a0 + kk);         // ds_load_b64, immediate offset
        v2f x1 = *(const v2f*)(a1 + kk);
        acc0 = __builtin_amdgcn_wmma_f32_16x16x4_f32(
            false, x0, false, b, (short)0, acc0, false, false);
        acc1 = __builtin_amdgcn_wmma_f32_16x16x4_f32(
            false, x1, false, b, (short)0, acc1, false, false);
    }

    float* __restrict__ ob0 = obase + out_off;
    float* __restrict__ ob1 = ob0 + (size_t)16 * OUT_DIM;
#pragma unroll
    for (int r = 0; r < 8; ++r) {
        if (!TAIL || r < rows_from_half) {
            ob0[(size_t)r * OUT_DIM] = acc0[r];
        }
    }
#pragma unroll
    for (int r = 0; r < 8; ++r) {
        if (!TAIL || r < rows_from_half - 16) {
            ob1[(size_t)r * OUT_DIM] = acc1[r];
        }
    }
}

template <bool TAIL>
__device__ __forceinline__ void run_tiles(const float* __restrict__ xsrow,
                                          const float* __restrict__ wbase,
                                          float* __restrict__ obase,
                                          int wave, int koff, int rows_from_half) {
    // ---- K = 128 : blocks g0,g1 (mats 0,1), 8 col-tiles each -> t in [0,16) ----
    for (int t = wave; t < 16; t += WAVES) {
        const int blk = t >> 3, c = t & 7;
        const int in_off = blk * 128;
        do_tile<128, TAIL>(xsrow, wbase, obase,
                           in_off, in_off + c * 16, blk * 16384 + c * 16,
                           koff, rows_from_half);
    }
    // ---- K = 64 : 6 blocks (mats 2,2,2,3,3,3), 4 col-tiles each -> t in [0,24) ----
    for (int t = wave; t < 24; t += WAVES) {
        const int blk = t >> 2, c = t & 3;
        const int in_off = 256 + blk * 64;
        const int mat = 2 + (blk >= 3 ? 1 : 0);
        do_tile<64, TAIL>(xsrow, wbase, obase,
                          in_off, in_off + c * 16, mat * 16384 + c * 16,
                          koff, rows_from_half);
    }
    // ---- K = 32 : 10 blocks (mats 4,5), 2 col-tiles each -> t in [0,20) ----
    for (int t = wave; t < 20; t += WAVES) {
        const int blk = t >> 1, c = t & 1;
        const int in_off = 640 + blk * 32;
        const int mat = 4 + (blk >= 5 ? 1 : 0);
        do_tile<32, TAIL>(xsrow, wbase, obase,
                          in_off, in_off + c * 16, mat * 16384 + c * 16,
                          koff, rows_from_half);
    }
    // ---- K = 16 : 14 blocks (mats 6,7), 1 col-tile each -> t in [0,14) ----
    for (int t = wave; t < 14; t += WAVES) {
        const int in_off = 960 + t * 16;
        const int mat = 6 + (t >= 7 ? 1 : 0);
        do_tile<16, TAIL>(xsrow, wbase, obase,
                          in_off, in_off, mat * 16384,
                          koff, rows_from_half);
    }
}

__global__ __launch_bounds__(WAVES * 32)
void IELin_block_diag_wmma(const float* __restrict__ x,
                           const float* __restrict__ W,
                           float* __restrict__ out,
                           int n) {
    extern __shared__ float xs[];

    const int tid  = threadIdx.x;
    const int row0 = blockIdx.x * ROWS;
    const bool tail = (row0 + ROWS > n);

    const int wave = __builtin_amdgcn_readfirstlane(tid >> 5);  // scalar wave id
    const int lane = tid & 31;

    // ---- Stage the 32-row x slab into LDS: each wave owns 4 rows ----
#pragma unroll
    for (int rr = 0; rr < 4; ++rr) {
        const int r = wave * 4 + rr;
        const int slot = (r & 15) * LDS_PAIR + (r >> 4) * LDS_HALF;
        const float4* __restrict__ src = (const float4*)(x + (size_t)(row0 + r) * IN_DIM);
        const bool rowok = !tail || (row0 + r < n);
        for (int j = lane; j < IN_DIM / 4; j += 32) {
            float4 v = rowok ? src[j] : make_float4(0.f, 0.f, 0.f, 0.f);
            *(float4*)&xs[slot + j * 4] = v;    // ds_store_b128
        }
    }
    __syncthreads();

    const int arow = lane & 15;        // A: M row held by this lane
    const int koff = (lane >> 4) << 1; // lanes 16-31 hold K+2,K+3
    const int bcol = lane & 15;        // B/D: N column held by this lane
    const int half = lane >> 4;        // D: lanes 16-31 hold M = 8..15

    const float* xsrow = &xs[arow * LDS_PAIR + koff];
    const float* wbase = W + bcol;
    float* obase = out + (size_t)(row0 + half * 8) * OUT_DIM + bcol;

    if (!tail) {
        run_tiles<false>(xsrow, wbase, obase, wave, koff, 32);
    } else {
        const int rows_here = n - row0;
        const int rows_from_half = rows_here - half * 8; // sub0: r < rfh; sub1: r < rfh-16
        run_tiles<true>(xsrow, wbase, obase, wave, koff, rows_from_half);
    }
}

extern "C" void kernel_launch(void* const* d_in, const int* in_sizes, int n_in,
                              void* d_out, int out_size, void* d_ws, size_t ws_size,
                              hipStream_t stream) {
    (void)n_in; (void)d_ws; (void)ws_size; (void)out_size;
    const float* x = (const float*)d_in[0];  // [N, 1184] f32
    const float* W = (const float*)d_in[1];  // [8, 128, 128] f32
    float* out = (float*)d_out;              // [N, 1184] f32
    const int n = in_sizes[0] / IN_DIM;

    const int grid = (n + ROWS - 1) / ROWS;
    IELin_block_diag_wmma<<<grid, WAVES * 32, LDS_BYTES, stream>>>(x, W, out, n);
}